// GraphEncoder_50173807952427
// MI455X (gfx1250) — compile-verified
//
#include <hip/hip_runtime.h>

typedef __attribute__((ext_vector_type(2))) float v2f;
typedef __attribute__((ext_vector_type(8))) float v8f;

#define IN_DIM 128
#define OUT_DIM 64
#define ROWS_PER_BLOCK 32
#define LDS_STRIDE 132  // 128 + 4 floats pad -> conflict-free ds_load_b64 across all 32 lanes

// ---------------------------------------------------------------------------
// Dense GEMM C[M,N] = A[M,K] * B[K,N] with V_WMMA_F32_16X16X4_F32.
//
// Block = (32, NCOLS/16) waves, covering a 32-row x NCOLS tile of C.
//   - The 32xK fp32 A stripe is staged ONCE per block into padded LDS via the
//     CDNA5 async global->LDS path (GLOBAL_LOAD_ASYNC_TO_LDS_B128, ASYNCcnt),
//     killing the 8x redundant per-wave global reads of A.
//   - Each wave owns one 16-column tile and BOTH row tiles: every B fragment
//     feeds two WMMAs, and the two accumulator chains are independent
//     (pipelines across the WMMA RAW-hazard window).
//
// A-operand (16x4 fp32) lane layout per ISA 7.12.2:
//   lanes 0-15 : M=lane,    {K+0,K+1};  lanes 16-31: M=lane-16, {K+2,K+3}
// C/D (16x16 f32): vreg r -> row r (lanes 0-15) / row r+8 (lanes 16-31).
// ---------------------------------------------------------------------------
template <int K, int NCOLS>
__global__ __launch_bounds__(32 * (NCOLS / 16))
void gemm_wmma_f32(const float* __restrict__ A,
                   const float* __restrict__ B,
                   float* __restrict__ C)
{
    __shared__ __align__(16) float sA[ROWS_PER_BLOCK * LDS_STRIDE];

    const int tid  = threadIdx.y * 32 + threadIdx.x;
    const int nthr = 32 * (NCOLS / 16);
    const int row0 = blockIdx.x * ROWS_PER_BLOCK;

    // ---- async-stage the 32xK A stripe into padded LDS ----
    {
        constexpr int chunksPerRow = (K * 4) / 16;                 // 16B chunks
        constexpr int totalChunks  = ROWS_PER_BLOCK * chunksPerRow;
        const unsigned sbase = (unsigned)(uintptr_t)sA;
#pragma unroll
        for (int i = 0; i < totalChunks / nthr; ++i) {
            int cidx = tid + i * nthr;
            int r    = cidx / chunksPerRow;
            int cc   = cidx % chunksPerRow;
            unsigned lds = sbase + (unsigned)(r * LDS_STRIDE + cc * 4) * 4u;
            unsigned long long g =
                (unsigned long long)(uintptr_t)(A + (size_t)(row0 + r) * K + cc * 4);
            asm volatile("global_load_async_to_lds_b128 %0, %1, off"
                         :: "v"(lds), "v"(g) : "memory");
        }
        asm volatile("s_wait_asynccnt 0x0" ::: "memory");
        __syncthreads();
    }

    const int lane = threadIdx.x;
    const int tN   = threadIdx.y;
    const int m    = lane & 15;
    const int half = lane >> 4;
    const int kh   = half * 2;
    const int col  = tN * 16 + m;

    const float* sA0 = sA + (size_t)m * LDS_STRIDE;         // row tile 0
    const float* sA1 = sA0 + 16 * LDS_STRIDE;               // row tile 1

    v8f acc0 = {}, acc1 = {};
#pragma unroll 4
    for (int k = 0; k < K; k += 4) {
        v2f a0 = *(const v2f*)(sA0 + k + kh);               // ds_load_b64, no conflicts
        v2f a1 = *(const v2f*)(sA1 + k + kh);
        v2f b;
        b.x = B[(size_t)(k + kh)     * NCOLS + col];
        b.y = B[(size_t)(k + kh + 1) * NCOLS + col];
        acc0 = __builtin_amdgcn_wmma_f32_16x16x4_f32(false, a0, false, b,
                                                     (short)0, acc0, false, false);
        acc1 = __builtin_amdgcn_wmma_f32_16x16x4_f32(false, a1, false, b,
                                                     (short)0, acc1, false, false);
    }

    float* c0 = C + (size_t)(row0 + half * 8) * NCOLS + col;
    float* c1 = c0 + (size_t)16 * NCOLS;
#pragma unroll
    for (int r = 0; r < 8; ++r) {
        c0[(size_t)r * NCOLS] = acc0[r];
        c1[(size_t)r * NCOLS] = acc1[r];
    }
}

// ---------------------------------------------------------------------------
// Edge-parallel SpMM scatter: acc[dst] += w_e * h[src].
// One wave covers one edge row (D=128) / half-wave per edge (D=64): coalesced
// float4 gathers of h (L2-resident: h fits in the 192MB L2) and hardware
// global_atomic_add_f32 that resolves at L2. Edge index/weight are streamed
// exactly once -> non-temporal loads so they never evict h from L2.
// ---------------------------------------------------------------------------
template <int D>
__global__ void spmm_scatter(const long long* __restrict__ ei,   // [2,E] int64
                             const float* __restrict__ ew,       // [E]
                             const float* __restrict__ h,        // [N,D]
                             float* __restrict__ acc,            // [N,D]
                             int E)
{
    constexpr int CH = D / 4;  // float4 chunks per edge row
    long long t = (long long)blockIdx.x * blockDim.x + threadIdx.x;
    int e = (int)(t / CH);
    int c = (int)(t % CH);
    if (e >= E) return;

    int   s = (int)__builtin_nontemporal_load(ei + e);
    int   d = (int)__builtin_nontemporal_load(ei + (size_t)E + e);
    float w = __builtin_nontemporal_load(ew + e);

    float4 v = ((const float4*)(h + (size_t)s * D))[c];
    float* o = acc + (size_t)d * D + (size_t)c * 4;
    unsafeAtomicAdd(o + 0, v.x * w);
    unsafeAtomicAdd(o + 1, v.y * w);
    unsafeAtomicAdd(o + 2, v.z * w);
    unsafeAtomicAdd(o + 3, v.w * w);
}

__global__ void zero_f4(float4* __restrict__ p, int n4)
{
    int i = blockIdx.x * blockDim.x + threadIdx.x;
    if (i < n4) p[i] = make_float4(0.f, 0.f, 0.f, 0.f);
}

// out = relu(acc + b) vectorized; D power of two
template <int D>
__global__ void bias_relu_f4(const float4* __restrict__ acc,
                             const float4* __restrict__ b,
                             float4* __restrict__ out, int n4)
{
    int i = blockIdx.x * blockDim.x + threadIdx.x;
    if (i >= n4) return;
    float4 v  = acc[i];
    float4 bb = b[i & (D / 4 - 1)];
    v.x = fmaxf(v.x + bb.x, 0.f);
    v.y = fmaxf(v.y + bb.y, 0.f);
    v.z = fmaxf(v.z + bb.z, 0.f);
    v.w = fmaxf(v.w + bb.w, 0.f);
    out[i] = v;
}

// out += b vectorized (final bias, no relu)
template <int D>
__global__ void bias_add_f4(float4* __restrict__ out,
                            const float4* __restrict__ b, int n4)
{
    int i = blockIdx.x * blockDim.x + threadIdx.x;
    if (i >= n4) return;
    float4 v  = out[i];
    float4 bb = b[i & (D / 4 - 1)];
    v.x += bb.x; v.y += bb.y; v.z += bb.z; v.w += bb.w;
    out[i] = v;
}

extern "C" void kernel_launch(void* const* d_in, const int* in_sizes, int n_in,
                              void* d_out, int out_size, void* d_ws, size_t ws_size,
                              hipStream_t stream)
{
    const float*     x  = (const float*)d_in[0];      // [N,128]
    const long long* ei = (const long long*)d_in[1];  // [2,E] int64
    const float*     ew = (const float*)d_in[2];      // [E]
    const float*     W1 = (const float*)d_in[3];      // [128,128]
    const float*     b1 = (const float*)d_in[4];      // [128]
    const float*     W2 = (const float*)d_in[5];      // [128,64]
    const float*     b2 = (const float*)d_in[6];      // [64]
    float*           out = (float*)d_out;             // [N,64]

    const int N = in_sizes[0] / IN_DIM;   // 100000 (multiple of 32)
    const int E = in_sizes[1] / 2;        // 600000

    // Workspace: h0 [N*128] | acc1 [N*128]; h1 reuses h0, h2 reuses acc1.
    float* h0   = (float*)d_ws;
    float* acc1 = h0 + (size_t)N * IN_DIM;
    float* h1   = h0;    // bias_relu(acc1) overwrites dead h0
    float* h2   = acc1;  // gemm2 output overwrites dead acc1

    const int n1 = N * IN_DIM;   // 12.8M
    const int n2 = N * OUT_DIM;  // 6.4M

    // ---- Layer 1 ----
    gemm_wmma_f32<IN_DIM, IN_DIM>
        <<<dim3(N / ROWS_PER_BLOCK), dim3(32, IN_DIM / 16), 0, stream>>>(x, W1, h0);

    zero_f4<<<(n1 / 4 + 255) / 256, 256, 0, stream>>>((float4*)acc1, n1 / 4);

    {
        long long threads = (long long)E * (IN_DIM / 4);
        spmm_scatter<IN_DIM>
            <<<(unsigned)((threads + 255) / 256), 256, 0, stream>>>(ei, ew, h0, acc1, E);
    }

    bias_relu_f4<IN_DIM><<<(n1 / 4 + 255) / 256, 256, 0, stream>>>(
        (const float4*)acc1, (const float4*)b1, (float4*)h1, n1 / 4);

    // ---- Layer 2 ----
    gemm_wmma_f32<IN_DIM, OUT_DIM>
        <<<dim3(N / ROWS_PER_BLOCK), dim3(32, OUT_DIM / 16), 0, stream>>>(h1, W2, h2);

    zero_f4<<<(n2 / 4 + 255) / 256, 256, 0, stream>>>((float4*)out, n2 / 4);

    {
        long long threads = (long long)E * (OUT_DIM / 4);
        spmm_scatter<OUT_DIM>
            <<<(unsigned)((threads + 255) / 256), 256, 0, stream>>>(ei, ew, h2, out, E);
    }

    bias_add_f4<OUT_DIM><<<(n2 / 4 + 255) / 256, 256, 0, stream>>>(
        (float4*)out, (const float4*)b2, n2 / 4);
}